// Non_local_47665547051237
// MI455X (gfx1250) — compile-verified
//
#include <hip/hip_runtime.h>
#include <math.h>

typedef float v2f __attribute__((ext_vector_type(2)));
typedef float v8f __attribute__((ext_vector_type(8)));

#define BATCH 8
#define CCH   512
#define NN    2048      // H*W = 64*32
#define SEMID 32

// ---------------------------------------------------------------------------
// Kernel 0: pack projection weights into wmat[512][16]: col0=g_w, col1=theta_w,
// col2=phi_w, cols 3..15 = 0.  This is the B-operand source for the WMMA GEMM.
// ---------------------------------------------------------------------------
__global__ __launch_bounds__(256) void wmat_kernel(
    const float* __restrict__ g_w, const float* __restrict__ th_w,
    const float* __restrict__ ph_w, float* __restrict__ wmat)
{
    int i = blockIdx.x * 256 + threadIdx.x;   // 512*16 = 8192 total
    if (i >= CCH * 16) return;
    int c = i >> 4, col = i & 15;
    float v = 0.0f;
    if (col == 0) v = g_w[c];
    else if (col == 1) v = th_w[c];
    else if (col == 2) v = ph_w[c];
    wmat[i] = v;
}

// ---------------------------------------------------------------------------
// Kernel 1: projections via V_WMMA_F32_16X16X4_F32.
// out[n, col] = sum_c x[b,c,n] * wmat[c,col],  cols 0..2 used (g, theta, phi).
// One wave per 16-row spatial tile; 128 WMMA steps over C=512 (K=4 each).
// A layout (f32 16x4): lanes 0-15 -> K=0(v0),1(v1); lanes 16-31 -> K=2,3.
// B layout (f32 4x16): N=lane%15? N=lane&15; same K split across lane halves.
// D layout (f32 16x16): vgpr r -> M=r (+8 for lanes>=16), N=lane&15.
// ---------------------------------------------------------------------------
__global__ __launch_bounds__(256) void proj_wmma_kernel(
    const float* __restrict__ x, const float* __restrict__ wmat,
    float* __restrict__ pg, float* __restrict__ pt, float* __restrict__ pp,
    const float* __restrict__ g_b, const float* __restrict__ th_b,
    const float* __restrict__ ph_b)
{
    int tid   = threadIdx.x;
    int lane  = tid & 31;
    int wv    = tid >> 5;
    int tile  = blockIdx.x * 8 + wv;         // 1024 tiles total
    int b     = tile >> 7;                   // N/16 = 128 tiles per batch
    int n0    = (tile & 127) << 4;
    int m     = lane & 15;                   // A row / B col within tile
    int khalf = (lane >> 4) << 1;            // 0 or 2

    const float* xb = x + (size_t)b * CCH * NN;
    v8f acc = {};
    #pragma unroll 4
    for (int k = 0; k < 128; ++k) {
        int c0 = (k << 2) + khalf;
        v2f a, bm;
        a.x  = xb[(size_t)c0 * NN + n0 + m];
        a.y  = xb[(size_t)(c0 + 1) * NN + n0 + m];
        bm.x = wmat[c0 * 16 + m];
        bm.y = wmat[(c0 + 1) * 16 + m];
        acc = __builtin_amdgcn_wmma_f32_16x16x4_f32(
                  false, a, false, bm, (short)0, acc, false, false);
    }

    int ncol = m;
    int mhi  = (lane >> 4) << 3;             // 0 or 8
    if (ncol < 3) {
        float bias  = (ncol == 0) ? g_b[0] : (ncol == 1 ? th_b[0] : ph_b[0]);
        float* dst  = (ncol == 0) ? pg : (ncol == 1 ? pt : pp);
        #pragma unroll
        for (int r = 0; r < 8; ++r)
            dst[b * NN + n0 + mhi + r] = acc[r] + bias;
    }
}

// ---------------------------------------------------------------------------
// Kernel 2: global average pool, one block per (b,c) row of 2048 floats.
// ---------------------------------------------------------------------------
__global__ __launch_bounds__(256) void avg_kernel(
    const float* __restrict__ x, float* __restrict__ xavg)
{
    __shared__ float sm[256];
    int bc = blockIdx.x;                     // 0..4095
    const float* row = x + (size_t)bc * NN;
    float s = 0.0f;
    for (int i = threadIdx.x; i < NN; i += 256) s += row[i];
    sm[threadIdx.x] = s;
    __syncthreads();
    for (int off = 128; off > 0; off >>= 1) {
        if (threadIdx.x < off) sm[threadIdx.x] += sm[threadIdx.x + off];
        __syncthreads();
    }
    if (threadIdx.x == 0) xavg[bc] = sm[0] * (1.0f / NN);
}

// ---------------------------------------------------------------------------
// Kernel 3: SE gate (se1 -> relu -> se2 -> sigmoid) + phi min/max per batch.
// One block per batch sample.
// ---------------------------------------------------------------------------
__global__ __launch_bounds__(256) void se_kernel(
    const float* __restrict__ xavg, const float* __restrict__ se1_w,
    const float* __restrict__ se1_b, const float* __restrict__ se2_w,
    const float* __restrict__ se2_b, const float* __restrict__ pp,
    float* __restrict__ wcw, float* __restrict__ phimax,
    float* __restrict__ phimin)
{
    __shared__ float savg[CCH];
    __shared__ float smid[SEMID];
    __shared__ float smax[256];
    __shared__ float smin[256];
    int b = blockIdx.x, tid = threadIdx.x;

    savg[tid]       = xavg[b * CCH + tid];
    savg[tid + 256] = xavg[b * CCH + tid + 256];
    __syncthreads();

    if (tid < SEMID) {
        float s = se1_b[tid];
        for (int c = 0; c < CCH; ++c) s += se1_w[tid * CCH + c] * savg[c];
        smid[tid] = s > 0.0f ? s : 0.0f;
    }
    __syncthreads();

    for (int c = tid; c < CCH; c += 256) {
        float s = se2_b[c];
        #pragma unroll
        for (int k = 0; k < SEMID; ++k) s += se2_w[c * SEMID + k] * smid[k];
        wcw[b * CCH + c] = 1.0f / (1.0f + __expf(-s));
    }

    const float* ph = pp + b * NN;
    float mx = -3.4e38f, mn = 3.4e38f;
    for (int i = tid; i < NN; i += 256) {
        float v = ph[i];
        mx = fmaxf(mx, v);
        mn = fminf(mn, v);
    }
    smax[tid] = mx; smin[tid] = mn;
    __syncthreads();
    for (int off = 128; off > 0; off >>= 1) {
        if (tid < off) {
            smax[tid] = fmaxf(smax[tid], smax[tid + off]);
            smin[tid] = fminf(smin[tid], smin[tid + off]);
        }
        __syncthreads();
    }
    if (tid == 0) { phimax[b] = smax[0]; phimin[b] = smin[0]; }
}

// ---------------------------------------------------------------------------
// Kernel 4: attention. f = outer(theta, phi) is rank-1, so the row softmax
// needs only one pass: m_i = t_i>=0 ? t_i*max(phi) : t_i*min(phi);
// y_i = sum_j e^{t_i phi_j - m_i} g_j / sum_j e^{t_i phi_j - m_i}.
// phi and g staged in LDS; one row per thread; 8 row-chunks per batch.
// ---------------------------------------------------------------------------
__global__ __launch_bounds__(256) void attn_kernel(
    const float* __restrict__ pt, const float* __restrict__ pp,
    const float* __restrict__ pg, const float* __restrict__ phimax,
    const float* __restrict__ phimin, float* __restrict__ y)
{
    __shared__ float sphi[NN];
    __shared__ float sg[NN];
    int b     = blockIdx.x >> 3;
    int chunk = blockIdx.x & 7;
    int tid   = threadIdx.x;
    for (int i = tid; i < NN; i += 256) {
        sphi[i] = pp[b * NN + i];
        sg[i]   = pg[b * NN + i];
    }
    __syncthreads();

    int   row = chunk * 256 + tid;
    float t   = pt[b * NN + row];
    float m   = (t >= 0.0f) ? t * phimax[b] : t * phimin[b];
    float den = 0.0f, num = 0.0f;
    #pragma unroll 4
    for (int j = 0; j < NN; ++j) {
        float e = __expf(fmaf(t, sphi[j], -m));
        den += e;
        num  = fmaf(e, sg[j], num);
    }
    y[b * NN + row] = num / den;
}

// ---------------------------------------------------------------------------
// Kernel 5: epilogue.  out = wcw[b,c] * ((W_w[c]*y + W_b[c] - mean)*inv + beta)
//                           + x, folded to out = A1[b,c]*y + A0[b,c] + x.
// float4 streaming.
// ---------------------------------------------------------------------------
__global__ __launch_bounds__(256) void final_kernel(
    const float* __restrict__ x, const float* __restrict__ y,
    const float* __restrict__ wcw, const float* __restrict__ W_w,
    const float* __restrict__ W_b, const float* __restrict__ gamma,
    const float* __restrict__ beta, const float* __restrict__ mean,
    const float* __restrict__ var, float* __restrict__ out)
{
    const int total4 = BATCH * CCH * NN / 4;      // 2,097,152
    for (int i4 = blockIdx.x * 256 + threadIdx.x; i4 < total4;
         i4 += gridDim.x * 256) {
        int n4 = i4 & 511;                        // NN/4 = 512
        int c  = (i4 >> 9) & 511;
        int b  = i4 >> 18;
        float inv = gamma[c] * rsqrtf(var[c] + 1e-5f);
        float gte = wcw[b * CCH + c];
        float A1  = gte * inv * W_w[c];
        float A0  = gte * fmaf(W_b[c] - mean[c], inv, beta[c]);
        float4 xv = ((const float4*)x)[i4];
        float4 yv = ((const float4*)(y + b * NN))[n4];
        float4 o;
        o.x = fmaf(A1, yv.x, A0) + xv.x;
        o.y = fmaf(A1, yv.y, A0) + xv.y;
        o.z = fmaf(A1, yv.z, A0) + xv.z;
        o.w = fmaf(A1, yv.w, A0) + xv.w;
        ((float4*)out)[i4] = o;
    }
}

// ---------------------------------------------------------------------------
extern "C" void kernel_launch(void* const* d_in, const int* in_sizes, int n_in,
                              void* d_out, int out_size, void* d_ws, size_t ws_size,
                              hipStream_t stream)
{
    const float* x        = (const float*)d_in[0];
    const float* g_w      = (const float*)d_in[1];
    const float* g_b      = (const float*)d_in[2];
    const float* theta_w  = (const float*)d_in[3];
    const float* theta_b  = (const float*)d_in[4];
    const float* phi_w    = (const float*)d_in[5];
    const float* phi_b    = (const float*)d_in[6];
    const float* W_w      = (const float*)d_in[7];
    const float* W_b      = (const float*)d_in[8];
    const float* bn_gamma = (const float*)d_in[9];
    const float* bn_beta  = (const float*)d_in[10];
    const float* bn_mean  = (const float*)d_in[11];
    const float* bn_var   = (const float*)d_in[12];
    const float* se1_w    = (const float*)d_in[13];
    const float* se1_b    = (const float*)d_in[14];
    const float* se2_w    = (const float*)d_in[15];
    const float* se2_b    = (const float*)d_in[16];
    float* out = (float*)d_out;

    char* ws = (char*)d_ws;
    size_t off = 0;
    float* wmat   = (float*)(ws + off); off += CCH * 16 * sizeof(float);      // 32 KB
    float* pg     = (float*)(ws + off); off += BATCH * NN * sizeof(float);    // 64 KB
    float* pt     = (float*)(ws + off); off += BATCH * NN * sizeof(float);
    float* pp     = (float*)(ws + off); off += BATCH * NN * sizeof(float);
    float* xavg   = (float*)(ws + off); off += BATCH * CCH * sizeof(float);   // 16 KB
    float* wcw    = (float*)(ws + off); off += BATCH * CCH * sizeof(float);
    float* phimax = (float*)(ws + off); off += 256;
    float* phimin = (float*)(ws + off); off += 256;
    float* yv     = (float*)(ws + off); off += BATCH * NN * sizeof(float);

    wmat_kernel<<<(CCH * 16 + 255) / 256, 256, 0, stream>>>(g_w, theta_w, phi_w, wmat);
    avg_kernel<<<BATCH * CCH, 256, 0, stream>>>(x, xavg);
    proj_wmma_kernel<<<BATCH * (NN / 16) / 8, 256, 0, stream>>>(
        x, wmat, pg, pt, pp, g_b, theta_b, phi_b);
    se_kernel<<<BATCH, 256, 0, stream>>>(xavg, se1_w, se1_b, se2_w, se2_b,
                                         pp, wcw, phimax, phimin);
    attn_kernel<<<BATCH * 8, 256, 0, stream>>>(pt, pp, pg, phimax, phimin, yv);
    final_kernel<<<4096, 256, 0, stream>>>(x, yv, wcw, W_w, W_b, bn_gamma,
                                           bn_beta, bn_mean, bn_var, out);
}